// AttentionAggregator_13537736917742
// MI455X (gfx1250) — compile-verified
//
#include <hip/hip_runtime.h>
#include <cstdint>

#define KN   10      // neighbors per node
#define DIM  128     // embedding dim
#define WPB  4       // waves per block
#define BLOCK (WPB * 32)
#define ROWF4 32     // float4 per 128-float row

typedef float v4f __attribute__((ext_vector_type(4)));

__global__ __launch_bounds__(BLOCK) void attn_agg_kernel(
    const float* __restrict__ table,
    const float* __restrict__ attn_w,
    const int*   __restrict__ nidx,
    float* __restrict__ out,
    int n_nodes)
{
    // per-wave double buffer: 2 x (10 rows x 32 lanes x float4) = 10240 B/wave
    __shared__ float4 smem[WPB * 2 * KN * ROWF4];

    const int lane        = threadIdx.x & 31;
    const int wid         = threadIdx.x >> 5;
    const int waves_total = gridDim.x * WPB;
    const int wave_id     = blockIdx.x * WPB + wid;

    if (wave_id >= n_nodes) return;   // wave-uniform guard, EXEC stays all-ones

    // this lane's slice of the attention weight vector
    const float4 w4 = reinterpret_cast<const float4*>(attn_w)[lane];

    const unsigned lds_base = (unsigned)(uintptr_t)&smem[0];
    const int buf_elems = KN * ROWF4;
    const int my0 = (wid * 2 + 0) * buf_elems;
    const int my1 = (wid * 2 + 1) * buf_elems;

    // Issue the K async gather loads (global -> LDS) for one node.
    // Indices are wave-uniform -> scalar s_loads; row data moves via
    // global_load_async_to_lds_b128 (ASYNCcnt-tracked, no dest VGPRs),
    // default RT temporal hint so rows stay resident in the 192MB L2.
    auto issue = [&](int unode, int bufsel) {
        const int* ip = nidx + (size_t)unode * KN;
        const int base = bufsel ? my1 : my0;
        #pragma unroll
        for (int k = 0; k < KN; ++k) {
            const int row = ip[k];  // uniform address -> s_load
            const float4* g =
                reinterpret_cast<const float4*>(table) + (size_t)row * ROWF4 + lane;
            const unsigned lds =
                lds_base + (unsigned)((base + k * ROWF4 + lane) * sizeof(float4));
            asm volatile("global_load_async_to_lds_b128 %0, %1, off"
                         :: "v"(lds), "v"(g)
                         : "memory");
        }
    };

    auto compute = [&](int node, int bufsel) {
        const int base = bufsel ? my1 : my0;
        float4 f[KN];
        float  s[KN];
        #pragma unroll
        for (int k = 0; k < KN; ++k) {
            f[k] = smem[base + k * ROWF4 + lane];                 // ds_load_b128
            s[k] = f[k].x * w4.x + f[k].y * w4.y + f[k].z * w4.z + f[k].w * w4.w;
        }
        // wave32 butterfly all-reduce of the K partial dot products
        #pragma unroll
        for (int off = 16; off > 0; off >>= 1) {
            #pragma unroll
            for (int k = 0; k < KN; ++k)
                s[k] += __shfl_xor(s[k], off, 32);
        }
        // softmax over K (redundant per lane; all lanes hold all K scores)
        float m = s[0];
        #pragma unroll
        for (int k = 1; k < KN; ++k) m = fmaxf(m, s[k]);
        float e[KN], sum = 0.f;
        #pragma unroll
        for (int k = 0; k < KN; ++k) { e[k] = __expf(s[k] - m); sum += e[k]; }
        const float inv = __builtin_amdgcn_rcpf(sum);             // v_rcp_f32
        v4f acc = (v4f)(0.f);
        #pragma unroll
        for (int k = 0; k < KN; ++k) {
            const float wk = e[k] * inv;
            acc.x += wk * f[k].x;
            acc.y += wk * f[k].y;
            acc.z += wk * f[k].z;
            acc.w += wk * f[k].w;
        }
        // Output is write-once / never re-read: non-temporal store keeps the
        // L2 dedicated to the embedding table (global_store_b128 th:NT).
        v4f* dst = reinterpret_cast<v4f*>(out + (size_t)node * DIM) + lane;
        __builtin_nontemporal_store(acc, dst);
    };

    int node = wave_id;
    int b = 0;
    issue(__builtin_amdgcn_readfirstlane(node), b);
    for (; node < n_nodes; node += waves_total) {
        const int nxt = node + waves_total;
        if (nxt < n_nodes) {
            // prefetch next node's rows into the other buffer, then wait for
            // the current buffer's 10 async loads (in-order completion).
            issue(__builtin_amdgcn_readfirstlane(nxt), b ^ 1);
            asm volatile("s_wait_asynccnt 0xa" ::: "memory");
        } else {
            asm volatile("s_wait_asynccnt 0x0" ::: "memory");
        }
        compute(node, b);
        b ^= 1;
    }
}

extern "C" void kernel_launch(void* const* d_in, const int* in_sizes, int n_in,
                              void* d_out, int out_size, void* d_ws, size_t ws_size,
                              hipStream_t stream) {
    const float* table  = (const float*)d_in[0];   // [VOCAB, 128] f32
    const float* attn_w = (const float*)d_in[1];   // [128] f32
    const int*   nidx   = (const int*)d_in[2];     // [N, 10] int
    float*       out    = (float*)d_out;           // [N, 128] f32

    const int n_nodes = in_sizes[2] / KN;          // 100000
    const int blocks  = 2048;                      // 8192 waves, ~12 nodes each

    attn_agg_kernel<<<blocks, BLOCK, 0, stream>>>(table, attn_w, nidx, out, n_nodes);
}